// MS_SSIM_Loss_25288767439586
// MI455X (gfx1250) — compile-verified
//
#include <hip/hip_runtime.h>
#include <hip/hip_bf16.h>
#include <math.h>

// MS-SSIM + L1 loss, MI455X (gfx1250), wave32.
// Separable 33-tap Gaussian conv as 16x48 * 48x16 Toeplitz matmuls using
// V_WMMA_F32_16X16X4_F32 (12 WMMAs per 16x16 tile per 1-D pass), full f32 accuracy.
// Horizontal stat pass keeps 5 accumulators live -> 5 straight-line WMMAs per
// K-step, no branching around matrix ops (EXEC stays all-ones).

typedef __attribute__((ext_vector_type(2))) float v2f;
typedef __attribute__((ext_vector_type(8))) float v8f;

#define IMG_H 512
#define IMG_W 512
#define PSTRIDE (IMG_H * IMG_W)     // one plane
#define NPLANES 38                  // 7 pairs * 5 stats + 3 L1 planes
#define NTILES 1024                 // 32x32 tiles of 16x16
#define C1_CONST 1.0e-4f
#define C2_CONST 9.0e-4f

// Toeplitz entry: T[K][N] = g[K - N] for 0 <= K-N <= 32 (33-tap filter, pad 16).
__device__ __forceinline__ float toep(const float* g, int kk, int n) {
    int t = kk - n;
    return (t >= 0 && t <= 32) ? g[t] : 0.0f;
}

__device__ __forceinline__ v8f wmma_f32x4(v2f a, v2f b, v8f c) {
    // (neg_a, A, neg_b, B, c_mod, C, reuse_a, reuse_b)
    return __builtin_amdgcn_wmma_f32_16x16x4_f32(false, a, false, b, (short)0, c,
                                                 false, false);
}

// ---------------------------------------------------------------------------
// Init: normalized 33-tap Gaussian for each sigma into ws.
// ---------------------------------------------------------------------------
__global__ void init_gtab_kernel(float* __restrict__ gtab) {
    int s = threadIdx.x;
    if (s < 5) {
        const float sigmas[5] = {0.5f, 1.0f, 2.0f, 4.0f, 8.0f};
        float tmp[33];
        float sum = 0.0f;
        float inv2s2 = 1.0f / (2.0f * sigmas[s] * sigmas[s]);
        for (int t = 0; t < 33; ++t) {
            float d = (float)(t - 16);
            float v = expf(-d * d * inv2s2);
            tmp[t] = v;
            sum += v;
        }
        float inv = 1.0f / sum;
        for (int t = 0; t < 33; ++t) gtab[s * 33 + t] = tmp[t] * inv;
    }
}

// ---------------------------------------------------------------------------
// Horizontal pass, statistic planes. One wave per (pair, tile).
// Pair j -> (channel, sigma); emits 5 planes: x, y, x^2, y^2, x*y.
// Output tile (16x16) = A(16x48 patch) * B(48x16 Toeplitz), 12 WMMAs/plane.
// ---------------------------------------------------------------------------
__global__ __launch_bounds__(256) void hpass_stat_kernel(
    const float* __restrict__ img1, const float* __restrict__ img2,
    float* __restrict__ planes, const float* __restrict__ gtab, int b) {
    __shared__ float lgt[5 * 33];
    __shared__ float patch[8][2][16 * 49];   // stride 49: bank-conflict free

    const int tid  = (int)threadIdx.x;
    const int wave = tid >> 5;
    const int lane = tid & 31;

    for (int i = tid; i < 5 * 33; i += 256) lgt[i] = gtab[i];

    const int wj   = (int)blockIdx.x * 8 + wave;     // [0, 7*1024)
    const int tile = wj & (NTILES - 1);
    const int pair = wj >> 10;                       // 0..6
    const int y0 = (tile >> 5) * 16;
    const int x0 = (tile & 31) * 16;

    const int pair_c[7] = {0, 0, 1, 1, 1, 2, 2};
    const int pair_s[7] = {0, 1, 1, 2, 3, 3, 4};
    const int c = pair_c[pair];
    const int s = pair_s[pair];

    const float* p1 = img1 + (size_t)(b * 3 + c) * PSTRIDE;
    const float* p2 = img2 + (size_t)(b * 3 + c) * PSTRIDE;
    float* sp1 = patch[wave][0];
    float* sp2 = patch[wave][1];

    // Stage 16x48 patches (cols x0-16 .. x0+31, zero-padded at edges).
    for (int idx = lane; idx < 16 * 48; idx += 32) {
        int r = idx / 48, cc = idx - r * 48;
        int gx = x0 - 16 + cc;
        bool ok = (gx >= 0) && (gx < IMG_W);
        size_t off = (size_t)(y0 + r) * IMG_W + gx;
        sp1[r * 49 + cc] = ok ? p1[off] : 0.0f;
        sp2[r * 49 + cc] = ok ? p2[off] : 0.0f;
    }
    __syncthreads();

    const float* g = lgt + s * 33;
    const int half = lane >> 4;          // K pair (2*half, 2*half+1) of each K-step
    const int n    = lane & 15;          // N for B/C; M for A

    // B fragments (Toeplitz filter), shared by all 5 statistic types.
    v2f bf[12];
#pragma unroll
    for (int k = 0; k < 12; ++k) {
        int kk = 4 * k + 2 * half;
        bf[k].x = toep(g, kk, n);
        bf[k].y = toep(g, kk + 1, n);
    }

    // A-layout: lane reads patch row (lane&15), cols 4k + 2*half + {0,1}.
    const float* a1 = sp1 + (lane & 15) * 49 + 2 * half;
    const float* a2 = sp2 + (lane & 15) * 49 + 2 * half;

    v8f accx = {}, accy = {}, accxx = {}, accyy = {}, accxy = {};
#pragma unroll
    for (int k = 0; k < 12; ++k) {
        float x1a = a1[4 * k], x1b = a1[4 * k + 1];
        float x2a = a2[4 * k], x2b = a2[4 * k + 1];
        v2f bfk = bf[k];
        v2f ax  = {x1a, x1b};
        v2f ay  = {x2a, x2b};
        v2f axx = {x1a * x1a, x1b * x1b};
        v2f ayy = {x2a * x2a, x2b * x2b};
        v2f axy = {x1a * x2a, x1b * x2b};
        accx  = wmma_f32x4(ax,  bfk, accx);
        accy  = wmma_f32x4(ay,  bfk, accy);
        accxx = wmma_f32x4(axx, bfk, accxx);
        accyy = wmma_f32x4(ayy, bfk, accyy);
        accxy = wmma_f32x4(axy, bfk, accxy);
    }

    // C/D layout: VGPR r -> row (r + 8*half), col n within the tile.
    size_t obase = (size_t)(y0 + 8 * half) * IMG_W + (x0 + n);
    float* o0 = planes + (size_t)(pair * 5 + 0) * PSTRIDE + obase;
    float* o1 = planes + (size_t)(pair * 5 + 1) * PSTRIDE + obase;
    float* o2 = planes + (size_t)(pair * 5 + 2) * PSTRIDE + obase;
    float* o3 = planes + (size_t)(pair * 5 + 3) * PSTRIDE + obase;
    float* o4 = planes + (size_t)(pair * 5 + 4) * PSTRIDE + obase;
#pragma unroll
    for (int r = 0; r < 8; ++r) {
        o0[(size_t)r * IMG_W] = accx[r];
        o1[(size_t)r * IMG_W] = accy[r];
        o2[(size_t)r * IMG_W] = accxx[r];
        o3[(size_t)r * IMG_W] = accyy[r];
        o4[(size_t)r * IMG_W] = accxy[r];
    }
}

// ---------------------------------------------------------------------------
// Horizontal pass, L1 planes: conv(|img1-img2|, g[sigma=8]) per channel.
// One wave per (channel, tile); planes 35..37.
// ---------------------------------------------------------------------------
__global__ __launch_bounds__(256) void hpass_l1_kernel(
    const float* __restrict__ img1, const float* __restrict__ img2,
    float* __restrict__ planes, const float* __restrict__ gtab, int b) {
    __shared__ float lgt[33];                 // sigma = 8 only
    __shared__ float patch[8][16 * 49];

    const int tid  = (int)threadIdx.x;
    const int wave = tid >> 5;
    const int lane = tid & 31;

    for (int i = tid; i < 33; i += 256) lgt[i] = gtab[4 * 33 + i];

    const int wj   = (int)blockIdx.x * 8 + wave;     // [0, 3*1024)
    const int tile = wj & (NTILES - 1);
    const int c    = wj >> 10;                       // 0..2
    const int y0 = (tile >> 5) * 16;
    const int x0 = (tile & 31) * 16;

    const float* p1 = img1 + (size_t)(b * 3 + c) * PSTRIDE;
    const float* p2 = img2 + (size_t)(b * 3 + c) * PSTRIDE;
    float* sp = patch[wave];

    for (int idx = lane; idx < 16 * 48; idx += 32) {
        int r = idx / 48, cc = idx - r * 48;
        int gx = x0 - 16 + cc;
        bool ok = (gx >= 0) && (gx < IMG_W);
        size_t off = (size_t)(y0 + r) * IMG_W + gx;
        sp[r * 49 + cc] = ok ? fabsf(p1[off] - p2[off]) : 0.0f;
    }
    __syncthreads();

    const int half = lane >> 4;
    const int n    = lane & 15;
    const float* ap = sp + (lane & 15) * 49 + 2 * half;

    v8f acc = {};
#pragma unroll
    for (int k = 0; k < 12; ++k) {
        int kk = 4 * k + 2 * half;
        v2f bfk;
        bfk.x = toep(lgt, kk, n);
        bfk.y = toep(lgt, kk + 1, n);
        v2f af = {ap[4 * k], ap[4 * k + 1]};
        acc = wmma_f32x4(af, bfk, acc);
    }

    float* outp = planes + (size_t)(35 + c) * PSTRIDE
                + (size_t)(y0 + 8 * half) * IMG_W + (x0 + n);
#pragma unroll
    for (int r = 0; r < 8; ++r) outp[(size_t)r * IMG_W] = acc[r];
}

// ---------------------------------------------------------------------------
// Vertical pass + fused SSIM math + tile reduction.
// Output tile (16x16) = A(16x48 Toeplitz) * B(48x16 H-plane patch).
// ---------------------------------------------------------------------------
__global__ __launch_bounds__(256) void vpass_kernel(
    const float* __restrict__ planes, const float* __restrict__ gtab,
    float* __restrict__ partial, int b) {
    __shared__ float lgt[5 * 33];
    __shared__ float vals[NPLANES * 256];
    __shared__ float red[256];

    const int tid  = (int)threadIdx.x;
    const int wave = tid >> 5;
    const int lane = tid & 31;
    for (int i = tid; i < 5 * 33; i += 256) lgt[i] = gtab[i];
    __syncthreads();

    const int tile = (int)blockIdx.x;
    const int y0 = (tile >> 5) * 16;
    const int x0 = (tile & 31) * 16;
    const int half = lane >> 4;
    const int n    = lane & 15;          // M (out row) for A; N (col) for B

    const int pair_s[7] = {0, 1, 1, 2, 3, 3, 4};

    for (int p = wave; p < NPLANES; p += 8) {   // wave-uniform trip count
        int s = (p < 35) ? pair_s[p / 5] : 4;
        const float* g = lgt + s * 33;
        const float* hp = planes + (size_t)p * PSTRIDE + (size_t)(x0 + n);
        v8f acc = {};
#pragma unroll
        for (int k = 0; k < 12; ++k) {
            int kk = 4 * k + 2 * half;
            v2f af;                               // A = Toeplitz(g): T[M][K]=g[K-M]
            af.x = toep(g, kk, n);
            af.y = toep(g, kk + 1, n);
            v2f bf;                               // B = H-plane rows (zero-padded)
            int r0 = y0 - 16 + kk;
            int r1 = r0 + 1;
            bf.x = (r0 >= 0 && r0 < IMG_H) ? hp[(size_t)r0 * IMG_W] : 0.0f;
            bf.y = (r1 >= 0 && r1 < IMG_H) ? hp[(size_t)r1 * IMG_W] : 0.0f;
            acc = wmma_f32x4(af, bf, acc);
        }
#pragma unroll
        for (int r = 0; r < 8; ++r)
            vals[p * 256 + (r + 8 * half) * 16 + n] = acc[r];
    }
    __syncthreads();

    // Per-pixel SSIM algebra; channel multiplicities for the 7 (c,sigma) pairs.
    const int mult[7] = {3, 2, 1, 3, 1, 2, 3};
    float pics = 1.0f;
    float lm = 0.0f;
#pragma unroll
    for (int q = 0; q < 7; ++q) {
        float mux = vals[(q * 5 + 0) * 256 + tid];
        float muy = vals[(q * 5 + 1) * 256 + tid];
        float ex2 = vals[(q * 5 + 2) * 256 + tid];
        float ey2 = vals[(q * 5 + 3) * 256 + tid];
        float exy = vals[(q * 5 + 4) * 256 + tid];
        float mux2 = mux * mux, muy2 = muy * muy, muxy = mux * muy;
        float cs = (2.0f * (exy - muxy) + C2_CONST) /
                   ((ex2 - mux2) + (ey2 - muy2) + C2_CONST);
        float csm = cs;
        if (mult[q] >= 2) csm *= cs;
        if (mult[q] == 3) csm *= cs;
        pics *= csm;
        if (q == 6) {
            float lc = (2.0f * muxy + C1_CONST) / (mux2 + muy2 + C1_CONST);
            lm = lc * lc * lc;
        }
    }
    float gl1 = (vals[35 * 256 + tid] + vals[36 * 256 + tid] + vals[37 * 256 + tid])
                * (1.0f / 3.0f);
    float loss = 0.025f * (1.0f - lm * pics) + 0.975f * gl1;

    red[tid] = loss;
    __syncthreads();
#pragma unroll
    for (int ofs = 128; ofs > 0; ofs >>= 1) {
        if (tid < ofs) red[tid] += red[tid + ofs];
        __syncthreads();
    }
    if (tid == 0) partial[b * NTILES + tile] = red[0];
}

// ---------------------------------------------------------------------------
// Final reduction: 8192 tile partials -> scalar 200 * mean.
// ---------------------------------------------------------------------------
__global__ __launch_bounds__(256) void reduce_kernel(const float* __restrict__ partial,
                                                     float* __restrict__ out) {
    __shared__ float red[256];
    int tid = (int)threadIdx.x;
    float s = 0.0f;
    for (int i = tid; i < 8 * NTILES; i += 256) s += partial[i];
    red[tid] = s;
    __syncthreads();
#pragma unroll
    for (int ofs = 128; ofs > 0; ofs >>= 1) {
        if (tid < ofs) red[tid] += red[tid + ofs];
        __syncthreads();
    }
    if (tid == 0) out[0] = red[0] * (200.0f / (8.0f * 512.0f * 512.0f));
}

extern "C" void kernel_launch(void* const* d_in, const int* in_sizes, int n_in,
                              void* d_out, int out_size, void* d_ws, size_t ws_size,
                              hipStream_t stream) {
    const float* img1 = (const float*)d_in[0];
    const float* img2 = (const float*)d_in[1];
    float* out = (float*)d_out;

    char* ws = (char*)d_ws;
    float* gtab    = (float*)ws;                       // 5*33 floats (pad to 1KB)
    float* partial = (float*)(ws + 1024);              // 8192 floats (32KB)
    float* planes  = (float*)(ws + 1024 + 32768);      // 38 * 512*512 floats (~38MB)

    init_gtab_kernel<<<1, 32, 0, stream>>>(gtab);
    for (int b = 0; b < 8; ++b) {
        hpass_stat_kernel<<<(NTILES * 7) / 8, 256, 0, stream>>>(img1, img2, planes,
                                                                gtab, b);
        hpass_l1_kernel<<<(NTILES * 3) / 8, 256, 0, stream>>>(img1, img2, planes,
                                                              gtab, b);
        vpass_kernel<<<NTILES, 256, 0, stream>>>(planes, gtab, partial, b);
    }
    reduce_kernel<<<1, 256, 0, stream>>>(partial, out);
}